// SAGEConv_77214922048102
// MI455X (gfx1250) — compile-verified
//
#include <hip/hip_runtime.h>

typedef float v2f __attribute__((ext_vector_type(2)));
typedef float v8f __attribute__((ext_vector_type(8)));

#define N_NODES 100000
#define N_EDGES 600000
#define D       128

// ---------------------------------------------------------------------------
// Kernel 1: zero the workspace (summed[N*D] followed by deg[N]).
// Runs every call so graph replay is deterministic.
// ---------------------------------------------------------------------------
__global__ void sage_zero_ws(float* __restrict__ ws, long long n) {
  long long i = (long long)blockIdx.x * blockDim.x + threadIdx.x;
  if (i < n) ws[i] = 0.0f;
}

// ---------------------------------------------------------------------------
// Kernel 2: edge gather + scatter-add.
// 128 threads cooperate on one edge: summed[dst][d] += feat[src][d].
// summed (51 MB) fits in the 192 MB L2, so the f32 atomics resolve in L2.
// ---------------------------------------------------------------------------
__global__ void sage_edge_scatter(const float* __restrict__ feat,
                                  const int*   __restrict__ src,
                                  const int*   __restrict__ dst,
                                  float*       __restrict__ summed,
                                  float*       __restrict__ deg) {
  long long idx = (long long)blockIdx.x * blockDim.x + threadIdx.x;
  long long e = idx >> 7;          // edge index
  int d = (int)(idx & 127);        // feature index
  if (e >= N_EDGES) return;
  int s = src[e];
  int t = dst[e];
  atomicAdd(&summed[(long long)t * D + d], feat[(long long)s * D + d]);
  if (d == 0) atomicAdd(&deg[t], 1.0f);
}

// ---------------------------------------------------------------------------
// Kernel 3: fused dual GEMM + bias via V_WMMA_F32_16X16X4_F32.
//   out = feat @ W_self^T + b_self + (summed/max(deg,1)) @ W_neigh^T + b_neigh
// One wave (32 threads) per 16-node x 16-outchannel tile.
//
// VGPR layouts (CDNA5 ISA, 32-bit operands, wave32):
//   A 16x4 : lane L -> row M = L%16; K = {0,1} for lanes 0..15, {2,3} for 16..31
//   B 4x16 : lane L -> col N = L%16; same K striping (col = output channel)
//   C/D    : VGPR r, lane L -> row M = r + 8*(L>=16), col N = L%16
// So every lane's A/B chunk is 2 consecutive floats -> b64 loads.
// ---------------------------------------------------------------------------
__global__ void sage_gemm_wmma(const float* __restrict__ feat,
                               const float* __restrict__ Ws,
                               const float* __restrict__ bs,
                               const float* __restrict__ Wn,
                               const float* __restrict__ bn,
                               const float* __restrict__ summed,
                               const float* __restrict__ deg,
                               float*       __restrict__ out) {
  const int lane = threadIdx.x;          // 0..31
  const int nb   = blockIdx.x * 16;      // node-row tile base
  const int ob   = blockIdx.y * 16;      // output-channel tile base
  const int m    = lane & 15;            // row (for A) / col (for B,C,D)
  const int hi   = lane >> 4;            // 0: K={0,1}   1: K={2,3}

  const float* arow  = feat   + (long long)(nb + m) * D + hi * 2; // A_self
  const float* srow  = summed + (long long)(nb + m) * D + hi * 2; // A_neigh (pre-scale)
  const float* wsrow = Ws     + (long long)(ob + m) * D + hi * 2; // B_self  col m
  const float* wnrow = Wn     + (long long)(ob + m) * D + hi * 2; // B_neigh col m

  const float dg  = deg[nb + m];
  const float inv = 1.0f / fmaxf(dg, 1.0f);   // mean; isolated nodes -> 0/1

  v8f c = {};   // f32 accumulator, 16x16 tile

#pragma unroll
  for (int k0 = 0; k0 < D; k0 += 4) {
    // self-path WMMA
    v2f a_s = *(const v2f*)(arow + k0);
    v2f b_s = *(const v2f*)(wsrow + k0);
    c = __builtin_amdgcn_wmma_f32_16x16x4_f32(
        /*neg_a=*/false, a_s, /*neg_b=*/false, b_s,
        /*c_mod=*/(short)0, c, /*reuse_a=*/false, /*reuse_b=*/false);

    // neighbor-path WMMA (apply mean normalization to A on the fly)
    v2f a_n = *(const v2f*)(srow + k0);
    a_n[0] *= inv;
    a_n[1] *= inv;
    v2f b_n = *(const v2f*)(wnrow + k0);
    c = __builtin_amdgcn_wmma_f32_16x16x4_f32(
        false, a_n, false, b_n, (short)0, c, false, false);
  }

  // D layout: VGPR r, this lane -> row nb + r + hi*8, col ob + m
  const float bias = bs[ob + m] + bn[ob + m];
#pragma unroll
  for (int r = 0; r < 8; ++r) {
    const int row = nb + r + hi * 8;
    out[(long long)row * D + (ob + m)] = c[r] + bias;
  }
}

// ---------------------------------------------------------------------------
// Host-side launcher. Inputs (setup_inputs order):
//   0: feat [N,128] f32   1: W_self [128,128] f32   2: b_self [128] f32
//   3: W_neigh [128,128]  4: b_neigh [128]          5: src [E] i32  6: dst [E] i32
// d_out: [N,128] f32. d_ws: summed[N*128] f32 + deg[N] f32 (51.6 MB).
// ---------------------------------------------------------------------------
extern "C" void kernel_launch(void* const* d_in, const int* in_sizes, int n_in,
                              void* d_out, int out_size, void* d_ws, size_t ws_size,
                              hipStream_t stream) {
  const float* feat = (const float*)d_in[0];
  const float* Ws   = (const float*)d_in[1];
  const float* bs   = (const float*)d_in[2];
  const float* Wn   = (const float*)d_in[3];
  const float* bn   = (const float*)d_in[4];
  const int*   src  = (const int*)d_in[5];
  const int*   dst  = (const int*)d_in[6];
  float*       out  = (float*)d_out;

  float* summed = (float*)d_ws;
  float* deg    = summed + (long long)N_NODES * D;

  // 1) zero summed + deg
  const long long nzero = (long long)N_NODES * D + N_NODES;
  {
    int block = 256;
    long long grid = (nzero + block - 1) / block;
    sage_zero_ws<<<dim3((unsigned)grid), dim3(block), 0, stream>>>(summed, nzero);
  }

  // 2) edge scatter-add (128 lanes per edge)
  {
    int block = 256;
    long long total = (long long)N_EDGES * D;
    long long grid = (total + block - 1) / block;
    sage_edge_scatter<<<dim3((unsigned)grid), dim3(block), 0, stream>>>(
        feat, src, dst, summed, deg);
  }

  // 3) fused dual GEMM + bias, one wave per 16x16 tile
  {
    dim3 grid(N_NODES / 16, D / 16);   // 6250 x 8
    dim3 block(32);                    // one wave32
    sage_gemm_wmma<<<grid, block, 0, stream>>>(feat, Ws, bs, Wn, bn,
                                               summed, deg, out);
  }
}